// DenseIouPred_42442866819104
// MI455X (gfx1250) — compile-verified
//
#include <hip/hip_runtime.h>
#include <stdint.h>

#define W72   72
#define NPIX  5184          // 72*72, elements per channel plane and output size
#define BLOCK 256

// Gather-form DenseIouPred:
//   out[h*72+w] = in-window && margins-valid ? IoU(pred(h,w), target+offsets) : 0
// Feature channels are staged global->LDS with CDNA5 async data-mover ops.
__global__ __launch_bounds__(BLOCK) void dense_iou_kernel(
    const float* __restrict__ output,    // (64,16,4,72,72); only first (4,72,72) used
    const int*   __restrict__ ind,       // only ind[0] used
    const float* __restrict__ target,    // only target[0..3] used
    const int*   __restrict__ radius_p,  // scalar radius (10)
    float*       __restrict__ iou_map)   // (72,72) float32
{
    __shared__ float smem[BLOCK * 4];

    const int tid = threadIdx.x;
    const int t   = blockIdx.x * BLOCK + tid;

    // ---- Stage this thread's 4 feature channels into LDS via async global->LDS ----
    // Clamp so the tail block issues in-range addresses (its results are unused).
    const int tc = (t < NPIX) ? t : (NPIX - 1);
    const unsigned off0 = (unsigned)(tc * 4);              // channel 0 byte offset
    const unsigned off1 = off0 + 1u * NPIX * 4u;           // +20736 B per channel
    const unsigned off2 = off0 + 2u * NPIX * 4u;
    const unsigned off3 = off0 + 3u * NPIX * 4u;
    // Low 32 bits of the generic (flat) shared-pointer == LDS byte address.
    const unsigned ldsa = (unsigned)(uintptr_t)(&smem[tid * 4]);

    asm volatile(
        "global_load_async_to_lds_b32 %0, %1, %8\n\t"
        "global_load_async_to_lds_b32 %2, %3, %8\n\t"
        "global_load_async_to_lds_b32 %4, %5, %8\n\t"
        "global_load_async_to_lds_b32 %6, %7, %8"
        :
        : "v"(ldsa),        "v"(off0),
          "v"(ldsa + 4u),   "v"(off1),
          "v"(ldsa + 8u),   "v"(off2),
          "v"(ldsa + 12u),  "v"(off3),
          "s"(output)
        : "memory");

    // Wait for this wave's async transfers; each thread reads only its own slots,
    // so no workgroup barrier is required.
    asm volatile("s_wait_asynccnt 0" ::: "memory");

    if (t < NPIX) {
        // Uniform scalars -> scalar loads.
        const int   ind0 = ind[0];
        const int   R    = radius_p[0];
        const float tg0  = target[0];
        const float tg1  = target[1];
        const float tg2  = target[2];
        const float tg3  = target[3];

        const int h  = t / W72;
        const int w  = t - h * W72;
        const int ch = ind0 / W72;
        const int cw = ind0 - ch * W72;
        const int rh = h - ch;
        const int rw = w - cw;

        float val = 0.0f;
        if (rh >= -R && rh <= R && rw >= -R && rw <= R) {
            const float frh  = (float)rh;
            const float frw  = (float)rw;
            const float t_wl = tg0 + frw;
            const float t_wr = tg1 - frw;
            const float t_ht = tg2 + frh;
            const float t_hb = tg3 - frh;
            // (h,w) is inside the 72x72 grid by construction, so the hh/ww range
            // checks of the reference are automatically satisfied here.
            if (t_wl >= 0.0f && t_wr >= 0.0f && t_ht >= 0.0f && t_hb >= 0.0f) {
                const float p_l = smem[tid * 4 + 0];
                const float p_r = smem[tid * 4 + 1];
                const float p_t = smem[tid * 4 + 2];
                const float p_b = smem[tid * 4 + 3];

                const float t_area = (t_wl + t_wr) * (t_ht + t_hb);
                const float p_area = (p_l + p_r) * (p_t + p_b);
                const float w_int  = fminf(p_l, t_wl) + fminf(p_r, t_wr);
                const float h_int  = fminf(p_b, t_hb) + fminf(p_t, t_ht);
                const float inter  = w_int * h_int;
                const float uni    = t_area + p_area - inter;
                val = (inter + 1.0f) / (uni + 1.0f);
            }
        }
        iou_map[t] = val;   // covers the zero-init of the map as well
    }
}

extern "C" void kernel_launch(void* const* d_in, const int* in_sizes, int n_in,
                              void* d_out, int out_size, void* d_ws, size_t ws_size,
                              hipStream_t stream) {
    (void)in_sizes; (void)n_in; (void)out_size; (void)d_ws; (void)ws_size;
    const float* output = (const float*)d_in[0];
    const int*   ind    = (const int*)d_in[1];
    const float* target = (const float*)d_in[2];
    const int*   radius = (const int*)d_in[3];
    float*       out    = (float*)d_out;

    const int grid = (NPIX + BLOCK - 1) / BLOCK;   // 21 blocks x 256 threads
    dense_iou_kernel<<<grid, BLOCK, 0, stream>>>(output, ind, target, radius, out);
}